// Lorenz96_17961553232130
// MI455X (gfx1250) — compile-verified
//
#include <hip/hip_runtime.h>
#include <cstdint>

// Lorenz-96 vector field, (B=262144, DIM=256) fp32. Memory-bound streaming
// kernel: one row per wave32, async global->LDS staging (gfx1250 path),
// double-buffered across rows, coalesced b128 loads/stores, NT stores.

typedef float v4f __attribute__((ext_vector_type(4)));

#define DIM 256
#define WAVES_PER_BLOCK 8
#define THREADS (WAVES_PER_BLOCK * 32)

// Issue async copy of one 1KB row (256 floats) into LDS.
// Lane l copies bytes [16l,16l+16) and [512+16l,512+16l+16): each b128 issue
// is fully coalesced across the wave (32 lanes x 16B = 512B contiguous).
__device__ __forceinline__ void async_load_row(const float* __restrict__ src,
                                               uint32_t lds_byte, int lane) {
  uint64_t g0 = (uint64_t)(uintptr_t)src + (uint32_t)(lane << 4);
  uint32_t l0 = lds_byte + (uint32_t)(lane << 4);
  asm volatile("global_load_async_to_lds_b128 %0, %1, off"
               :: "v"(l0), "v"(g0) : "memory");
  asm volatile("global_load_async_to_lds_b128 %0, %1, off"
               :: "v"(l0 + 512u), "v"(g0 + 512ull) : "memory");
}

// One Lorenz-96 element: p0*(s[c+1]-s[c-2])*s[c-1] - p1*s[c] + p2, wrap mod 256.
__device__ __forceinline__ float lz(const float* __restrict__ row, int c,
                                    float p0, float p1, float p2) {
  float sp1 = row[(c + 1) & (DIM - 1)];
  float sm2 = row[(c + DIM - 2) & (DIM - 1)];
  float sm1 = row[(c + DIM - 1) & (DIM - 1)];
  float s   = row[c & (DIM - 1)];
  return fmaf(p0 * (sp1 - sm2), sm1, fmaf(-p1, s, p2));
}

__global__ __launch_bounds__(THREADS)
void lorenz96_kernel(const float* __restrict__ state,
                     const float* __restrict__ params,
                     float* __restrict__ out, int batch) {
  // 2 row-buffers per wave (double buffer), 16B aligned for b128 async writes.
  __shared__ __align__(16) float smem[WAVES_PER_BLOCK * 2 * DIM];

  const int lane = (int)(threadIdx.x & 31u);
  const int wv   = (int)(threadIdx.x >> 5u);

  float* buf0 = &smem[(wv * 2 + 0) * DIM];
  float* buf1 = &smem[(wv * 2 + 1) * DIM];
  // Low 32 bits of the generic pointer to LDS == LDS byte address.
  const uint32_t lds0 = (uint32_t)(uintptr_t)buf0;
  const uint32_t lds1 = (uint32_t)(uintptr_t)buf1;

  // Lane l owns columns [4l,4l+4) and [128+4l,128+4l+4).
  const int c0 = lane << 2;
  const int c1 = 128 + (lane << 2);

  // Per-column params, loaded once (reused for all rows this wave handles).
  const v4f* __restrict__ P = (const v4f*)params;          // 3 x 64 x v4f
  const v4f p0a = P[lane],       p0b = P[32 + lane];
  const v4f p1a = P[64 + lane],  p1b = P[96 + lane];
  const v4f p2a = P[128 + lane], p2b = P[160 + lane];

  const int nWaves = (int)gridDim.x * WAVES_PER_BLOCK;     // wave-stride
  int b = (int)blockIdx.x * WAVES_PER_BLOCK + wv;
  if (b >= batch) return;

  async_load_row(state + (size_t)b * DIM, lds0, lane);     // prologue
  int cur = 0;
  for (; b < batch; b += nWaves) {
    const int bn = b + nWaves;
    const bool pf = bn < batch;
    if (pf) {
      // Prefetch next row into the other buffer, then wait for only the
      // current row's two loads (<=2 outstanding = the prefetch pair).
      async_load_row(state + (size_t)bn * DIM, cur ? lds0 : lds1, lane);
      asm volatile("s_wait_asynccnt 2" ::: "memory");
    } else {
      asm volatile("s_wait_asynccnt 0" ::: "memory");
    }

    const float* row = cur ? buf1 : buf0;
    float* dst = out + (size_t)b * DIM;

    v4f oa, ob;
    oa.x = lz(row, c0 + 0, p0a.x, p1a.x, p2a.x);
    oa.y = lz(row, c0 + 1, p0a.y, p1a.y, p2a.y);
    oa.z = lz(row, c0 + 2, p0a.z, p1a.z, p2a.z);
    oa.w = lz(row, c0 + 3, p0a.w, p1a.w, p2a.w);
    ob.x = lz(row, c1 + 0, p0b.x, p1b.x, p2b.x);
    ob.y = lz(row, c1 + 1, p0b.y, p1b.y, p2b.y);
    ob.z = lz(row, c1 + 2, p0b.z, p1b.z, p2b.z);
    ob.w = lz(row, c1 + 3, p0b.w, p1b.w, p2b.w);

    // Streamed once -> non-temporal b128 stores, fully coalesced.
    __builtin_nontemporal_store(oa, (v4f*)(dst + c0));
    __builtin_nontemporal_store(ob, (v4f*)(dst + c1));

    cur ^= 1;
  }
}

extern "C" void kernel_launch(void* const* d_in, const int* in_sizes, int n_in,
                              void* d_out, int out_size, void* d_ws, size_t ws_size,
                              hipStream_t stream) {
  const float* state  = (const float*)d_in[0];   // (B, 256) fp32
  const float* params = (const float*)d_in[1];   // (3, 256) fp32
  // d_in[2] = t, unused by the vector field.
  float* out = (float*)d_out;

  const int batch = in_sizes[0] / DIM;           // 262144
  // 8192 blocks x 8 waves = 65536 waves; 4 rows/wave at B=262144 so the
  // double-buffered pipeline has work to overlap. Stride loop handles any B.
  int grid = 8192;
  int maxBlocks = (batch + WAVES_PER_BLOCK - 1) / WAVES_PER_BLOCK;
  if (grid > maxBlocks) grid = maxBlocks;
  if (grid < 1) grid = 1;

  lorenz96_kernel<<<grid, THREADS, 0, stream>>>(state, params, out, batch);
}